// BloomAttention_43843026157714
// MI455X (gfx1250) — compile-verified
//
#include <hip/hip_runtime.h>
#include <math.h>

// ---------------------------------------------------------------------------
// BloomAttention for MI455X (gfx1250): fp32 WMMA (V_WMMA_F32_16X16X4_F32).
// Roofline: 268MB fp32 weights @ 23.3TB/s ~ 11.5us vs ~412 GFLOP (~1us on the
// matrix pipe) -> HBM-bound. Exact fp32 WMMA keeps reference numerics; the
// optimization effort goes into memory pipelining: register double-buffered
// LDS staging + global_prefetch_b8 lookahead.
// ---------------------------------------------------------------------------

typedef __attribute__((ext_vector_type(2))) float v2f;
typedef __attribute__((ext_vector_type(8))) float v8f;

#define S_LEN   2048
#define H_DIM   4096
#define NH      32
#define HD      128
#define QKV_N   12288
#define INV_NORM 0.08838834764831845f   // 1/sqrt(128)
#define NEG_MIN  (-3.4028234663852886e38f)

static __device__ __forceinline__ v8f wmma_f32(v2f a, v2f b, v8f c) {
    // 8 args: (neg_a, A, neg_b, B, c_mod, C, reuse_a, reuse_b)
    return __builtin_amdgcn_wmma_f32_16x16x4_f32(false, a, false, b,
                                                 (short)0, c, false, false);
}

// ---------------------------------------------------------------------------
// C[M,N] = A[M,K] * B[N,K]^T + bias[N] (+ res[M,N])   (row-major everywhere)
// Block: 256 threads = 8 waves, tile 32(M) x 64(N), K chunk 32.
// Register double-buffering: chunk k+1 global loads issue before chunk k's
// WMMA steps; prefetch warms GL2 two chunks ahead.
// ---------------------------------------------------------------------------
template <bool ADD_RES>
__global__ __launch_bounds__(256) void gemm_nt_wmma(
    const float* __restrict__ A, const float* __restrict__ Bw,
    const float* __restrict__ bias, const float* __restrict__ res,
    float* __restrict__ C, int M, int N, int K)
{
    __shared__ float As[32][36];
    __shared__ float Bs[64][36];

    const int tid  = threadIdx.x;
    const int lane = tid & 31;
    const int wav  = tid >> 5;
    const int half = lane >> 4;    // 0: lanes 0-15, 1: lanes 16-31
    const int lm   = lane & 15;

    const int wm = wav >> 2;       // 0..1  (M sub-tile)
    const int wn = wav & 3;        // 0..3  (N sub-tile)

    const int m0 = blockIdx.y * 32;
    const int n0 = blockIdx.x * 64;

    v8f acc = {};

    const int a_row = tid >> 3;            // 0..31
    const int a_c4  = (tid & 7) << 2;      // 0,4,..,28

    const float* aptr  = A  + (size_t)(m0 + a_row) * K + a_c4;
    const float* bptr0 = Bw + (size_t)(n0 + a_row) * K + a_c4;
    const float* bptr1 = Bw + (size_t)(n0 + a_row + 32) * K + a_c4;

    // prologue: stage chunk 0 into registers
    float4 ra  = *reinterpret_cast<const float4*>(aptr);
    float4 rb0 = *reinterpret_cast<const float4*>(bptr0);
    float4 rb1 = *reinterpret_cast<const float4*>(bptr1);

    for (int k0 = 0; k0 < K; k0 += 32) {
        // commit staged registers to LDS
        As[a_row][a_c4 + 0] = ra.x;  As[a_row][a_c4 + 1] = ra.y;
        As[a_row][a_c4 + 2] = ra.z;  As[a_row][a_c4 + 3] = ra.w;
        Bs[a_row][a_c4 + 0] = rb0.x; Bs[a_row][a_c4 + 1] = rb0.y;
        Bs[a_row][a_c4 + 2] = rb0.z; Bs[a_row][a_c4 + 3] = rb0.w;
        Bs[a_row + 32][a_c4 + 0] = rb1.x; Bs[a_row + 32][a_c4 + 1] = rb1.y;
        Bs[a_row + 32][a_c4 + 2] = rb1.z; Bs[a_row + 32][a_c4 + 3] = rb1.w;
        __syncthreads();

        // issue next chunk's global loads now; latency hides under WMMA below
        if (k0 + 32 < K) {
            ra  = *reinterpret_cast<const float4*>(aptr  + k0 + 32);
            rb0 = *reinterpret_cast<const float4*>(bptr0 + k0 + 32);
            rb1 = *reinterpret_cast<const float4*>(bptr1 + k0 + 32);
        }
        // warm GL2 two chunks ahead (global_prefetch_b8, speculative)
        if (k0 + 64 < K) {
            __builtin_prefetch(aptr  + k0 + 64, 0, 1);
            __builtin_prefetch(bptr0 + k0 + 64, 0, 1);
            __builtin_prefetch(bptr1 + k0 + 64, 0, 1);
        }

        #pragma unroll
        for (int kk = 0; kk < 32; kk += 4) {
            const int ka = kk + half * 2;
            v2f a, b;
            a.x = As[wm * 16 + lm][ka];
            a.y = As[wm * 16 + lm][ka + 1];
            b.x = Bs[wn * 16 + lm][ka];
            b.y = Bs[wn * 16 + lm][ka + 1];
            acc = wmma_f32(a, b, acc);
        }
        __syncthreads();
    }

    // epilogue: C layout -> VGPR r holds M = r + 8*half, N = lm
    const int col = n0 + wn * 16 + lm;
    const float bv = bias[col];
    #pragma unroll
    for (int r = 0; r < 8; ++r) {
        const int row = m0 + wm * 16 + (half ? r + 8 : r);
        float v = acc[r] + bv;
        if (ADD_RES) v += res[(size_t)row * N + col];
        C[(size_t)row * N + col] = v;
    }
}

// ---------------------------------------------------------------------------
// Flash attention, one block = (head, 32-query tile), 8 waves.
// fused layout per token: [h][3][128] -> q at h*384, k at h*384+128, v at +256
// ---------------------------------------------------------------------------
__device__ __forceinline__ void load_tile_32x128(
    float dst[32][132], const float* __restrict__ src,
    int row0, int off, int tid)
{
    #pragma unroll
    for (int i = 0; i < 4; ++i) {
        const int lin = tid + i * 256;       // float4 index, 0..1023
        const int r   = lin >> 5;            // 32 float4 per row
        const int c4  = (lin & 31) << 2;
        const float4 v = *reinterpret_cast<const float4*>(
            src + (size_t)(row0 + r) * QKV_N + off + c4);
        dst[r][c4 + 0] = v.x; dst[r][c4 + 1] = v.y;
        dst[r][c4 + 2] = v.z; dst[r][c4 + 3] = v.w;
    }
}

__global__ __launch_bounds__(256) void attn_kernel(
    const float* __restrict__ fused, const float* __restrict__ alibi,
    float* __restrict__ ctx)
{
    __shared__ float Qs[32][132];
    __shared__ float Ks[32][132];
    __shared__ float Vs[32][132];
    __shared__ float Ps[32][36];
    __shared__ float scaleS[32];
    __shared__ float lS[32];

    const int tid  = threadIdx.x;
    const int lane = tid & 31;
    const int wav  = tid >> 5;
    const int half = lane >> 4;
    const int lm   = lane & 15;

    const int qsub = wav & 1;      // which 16-row half of the 32-query tile
    const int nq   = wav >> 1;     // 0..3 -> 32 HD columns each

    const int h  = blockIdx.y;
    const int qb = blockIdx.x;
    const int q0 = qb * 32;
    const int qoff = h * 384;

    load_tile_32x128(Qs, fused, q0, qoff, tid);

    v8f acc0 = {}, acc1 = {};
    float mrow[8], lrow[8];
    #pragma unroll
    for (int r = 0; r < 8; ++r) { mrow[r] = -INFINITY; lrow[r] = 0.0f; }

    const int nkb = qb + 1;   // causal: key chunks 0..qb
    for (int kb = 0; kb < nkb; ++kb) {
        const int k0 = kb * 32;
        __syncthreads();   // previous iteration's P*V done before reloading K/V
        load_tile_32x128(Ks, fused, k0, qoff + 128, tid);
        load_tile_32x128(Vs, fused, k0, qoff + 256, tid);
        __syncthreads();

        // prefetch next key block's K/V rows into GL2 (speculative)
        if (kb + 1 < nkb) {
            const int pr = tid >> 3;                 // 0..31
            const int pc = (tid & 7) << 4;           // 16-float (64B) segments
            const float* base =
                fused + (size_t)(k0 + 32 + pr) * QKV_N + qoff + pc;
            __builtin_prefetch(base + 128, 0, 1);    // K row segment
            __builtin_prefetch(base + 256, 0, 1);    // V row segment
        }

        if (wav < 2) {     // wave 0 -> qsub 0, wave 1 -> qsub 1 (wave-uniform)
            // scores: S(16x32) = Q(16x128) * K^T(128x32), two 16x16 tiles
            v8f s0 = {}, s1 = {};
            #pragma unroll
            for (int kk = 0; kk < 128; kk += 4) {
                const int ka = kk + half * 2;
                v2f a, b0, b1;
                a.x  = Qs[qsub * 16 + lm][ka];  a.y  = Qs[qsub * 16 + lm][ka + 1];
                b0.x = Ks[lm][ka];              b0.y = Ks[lm][ka + 1];
                b1.x = Ks[16 + lm][ka];         b1.y = Ks[16 + lm][ka + 1];
                s0 = wmma_f32(a, b0, s0);
                s1 = wmma_f32(a, b1, s1);
            }
            const int key0 = k0 + lm;
            const int key1 = k0 + 16 + lm;
            const float al0 = alibi[h * S_LEN + key0];
            const float al1 = alibi[h * S_LEN + key1];
            #pragma unroll
            for (int r = 0; r < 8; ++r) {
                const int rowM  = half ? r + 8 : r;
                const int query = q0 + qsub * 16 + rowM;
                float v0 = s0[r] * INV_NORM + al0;
                float v1 = s1[r] * INV_NORM + al1;
                if (key0 > query) v0 = NEG_MIN;
                if (key1 > query) v1 = NEG_MIN;
                // row max over the 16-lane group (xor masks stay in-group)
                float mx = fmaxf(v0, v1);
                #pragma unroll
                for (int msk = 8; msk >= 1; msk >>= 1)
                    mx = fmaxf(mx, __shfl_xor(mx, msk, 32));
                const float mnew = fmaxf(mrow[r], mx);
                const float p0 = __expf(v0 - mnew);
                const float p1 = __expf(v1 - mnew);
                float sum = p0 + p1;
                #pragma unroll
                for (int msk = 8; msk >= 1; msk >>= 1)
                    sum += __shfl_xor(sum, msk, 32);
                const float sc = __expf(mrow[r] - mnew);
                lrow[r] = lrow[r] * sc + sum;
                mrow[r] = mnew;
                const int prow = qsub * 16 + rowM;
                Ps[prow][lm]      = p0;
                Ps[prow][16 + lm] = p1;
                if (lm == 0) scaleS[prow] = sc;
            }
        }
        __syncthreads();

        // all waves: rescale accumulators, then D += P(16x32) * V(32x32slice)
        #pragma unroll
        for (int r = 0; r < 8; ++r) {
            const float sc = scaleS[qsub * 16 + (half ? r + 8 : r)];
            acc0[r] *= sc;
            acc1[r] *= sc;
        }
        const int nc0 = nq * 32 + lm;
        const int nc1 = nq * 32 + 16 + lm;
        #pragma unroll
        for (int kk = 0; kk < 32; kk += 4) {
            const int ka = kk + half * 2;
            v2f a, b0, b1;
            a.x  = Ps[qsub * 16 + lm][ka];  a.y  = Ps[qsub * 16 + lm][ka + 1];
            b0.x = Vs[ka][nc0];             b0.y = Vs[ka + 1][nc0];
            b1.x = Vs[ka][nc1];             b1.y = Vs[ka + 1][nc1];
            acc0 = wmma_f32(a, b0, acc0);
            acc1 = wmma_f32(a, b1, acc1);
        }
    }

    if (wav < 2) {
        #pragma unroll
        for (int r = 0; r < 8; ++r)
            if (lm == 0) lS[qsub * 16 + (half ? r + 8 : r)] = lrow[r];
    }
    __syncthreads();

    #pragma unroll
    for (int r = 0; r < 8; ++r) {
        const int rowM = half ? r + 8 : r;
        const float inv = 1.0f / lS[qsub * 16 + rowM];
        const int row = q0 + qsub * 16 + rowM;
        const int col = h * HD + nq * 32 + lm;
        ctx[(size_t)row * H_DIM + col]      = acc0[r] * inv;
        ctx[(size_t)row * H_DIM + col + 16] = acc1[r] * inv;
    }
}

// ---------------------------------------------------------------------------
extern "C" void kernel_launch(void* const* d_in, const int* in_sizes, int n_in,
                              void* d_out, int out_size, void* d_ws, size_t ws_size,
                              hipStream_t stream) {
    const float* hidden   = (const float*)d_in[0];
    const float* residual = (const float*)d_in[1];
    const float* alibi    = (const float*)d_in[2];
    // d_in[3] = attention_mask (bool) — causal mask applied analytically
    const float* w_qkv    = (const float*)d_in[4];
    const float* b_qkv    = (const float*)d_in[5];
    const float* w_dense  = (const float*)d_in[6];
    const float* b_dense  = (const float*)d_in[7];
    float* out = (float*)d_out;

    float* fused = (float*)d_ws;                          // 2048 x 12288 f32
    float* ctx   = fused + (size_t)S_LEN * QKV_N;         // 2048 x 4096  f32

    dim3 blk(256);

    // 1) fused = hidden @ w_qkv^T + b_qkv
    gemm_nt_wmma<false><<<dim3(QKV_N / 64, S_LEN / 32), blk, 0, stream>>>(
        hidden, w_qkv, b_qkv, nullptr, fused, S_LEN, QKV_N, H_DIM);

    // 2) ctx = softmax(QK^T * inv_norm + alibi, causal) @ V
    attn_kernel<<<dim3(S_LEN / 32, NH), blk, 0, stream>>>(fused, alibi, ctx);

    // 3) out = ctx @ w_dense^T + b_dense + residual
    gemm_nt_wmma<true><<<dim3(H_DIM / 64, S_LEN / 32), blk, 0, stream>>>(
        ctx, w_dense, b_dense, residual, out, S_LEN, H_DIM, H_DIM);
}